// kmax_pooling_42528766165383
// MI455X (gfx1250) — compile-verified
//
#include <hip/hip_runtime.h>

// ---------------------------------------------------------------------------
// k-max pooling (K=8) over rows of 4096 f32, output in original index order.
// One wave32 per row. Row staged to LDS via CDNA5 async global->LDS loads
// (ASYNCcnt), pipelined with partial s_wait_asynccnt waits. The async loads'
// per-lane LDS destinations transpose the row so each lane's 128 elements are
// contiguous (XOR-swizzled against bank conflicts) -> all LDS reads are b128.
// Selection: per-lane chunk maxima + 8 rounds of wave-wide max extraction
// with exact lowest-index tie-breaking (matches jax.lax.top_k semantics).
// ---------------------------------------------------------------------------

constexpr int LROW   = 4096;   // row length
constexpr int KTOP   = 8;      // k-max
constexpr int WAVES  = 4;      // waves (rows) per block -> 64 KB LDS
constexpr int NCHUNK = 8;      // chunks per lane (16 elements = 4 float4 each)
constexpr int NGRP   = 32;     // float4 groups per lane

#if defined(__has_builtin)
#if __has_builtin(__builtin_amdgcn_global_load_async_to_lds_b128)
#define HAVE_ASYNC_LDS 1
#endif
#endif
#ifndef HAVE_ASYNC_LDS
#define HAVE_ASYNC_LDS 0
#endif

// Builtin signature (probed via hipcc diagnostics): (AS1 int4*, AS3 int4*, int, int)
typedef int v4i __attribute__((vector_size(16)));
typedef __attribute__((address_space(1))) v4i* gv4i_p;
typedef __attribute__((address_space(3))) v4i* lv4i_p;

#define NEG_INF (-__builtin_inff())

__global__ __launch_bounds__(WAVES * 32) void
kmax_pooling_kernel(const float* __restrict__ x, float* __restrict__ out, int nrows) {
  // 4096 float4 = 64 KB; float4 type guarantees 16B alignment for b128 LDS ops.
  __shared__ float4 tile4[WAVES * LROW / 4];

  const int lane = threadIdx.x & 31;
  const int wave = threadIdx.x >> 5;
  const int row  = blockIdx.x * WAVES + wave;
  if (row >= nrows) return;

  const float* gsrc = x + (size_t)row * LROW;
  float4* l4 = tile4 + wave * (LROW / 4);  // 1024 float4 per wave
  float*  lf = (float*)l4;

  // ---- Stage A: async global->LDS, transposing at store time --------------
  // Issue j: lanes fetch global float4s (j*32+lane); lane l's group j lands in
  // its own contiguous region at slot (j ^ l)  (XOR swizzle vs bank conflicts).
  // Lane l thus owns global elements e = 128j + 4l + t  (j=0..31, t=0..3).
#pragma unroll
  for (int j = 0; j < NGRP; ++j) {
    const int goff = (j * 32 + lane) * 4;        // float index in row
    const int ldst = lane * NGRP + (j ^ lane);   // float4 slot in wave region
#if HAVE_ASYNC_LDS
    __builtin_amdgcn_global_load_async_to_lds_b128(
        (gv4i_p)(gsrc + goff), (lv4i_p)(l4 + ldst), /*offset=*/0, /*cpol=*/0);
#else
    l4[ldst] = *(const float4*)(gsrc + goff);
#endif
  }

  // ---- Stage B: per-lane chunk maxima (4 x b128 per chunk) ----------------
  // Chunk k needs async issues j = 4k..4k+3; async loads complete in order,
  // so wait asynccnt <= 28-4k before chunk k (pipelines compute under DMA).
  float cmax[NCHUNK];

#define STAGE_B_CHUNK(kc)                                                     \
  do {                                                                        \
    asm volatile("s_wait_asynccnt %0" ::"i"(28 - 4 * (kc)) : "memory");       \
    float m_ = NEG_INF;                                                       \
    _Pragma("unroll") for (int g = 0; g < 4; ++g) {                           \
      const float4 q = l4[lane * NGRP + ((4 * (kc) + g) ^ lane)];             \
      m_ = fmaxf(m_, fmaxf(fmaxf(q.x, q.y), fmaxf(q.z, q.w)));                \
    }                                                                         \
    cmax[kc] = m_;                                                            \
  } while (0)

  STAGE_B_CHUNK(0); STAGE_B_CHUNK(1); STAGE_B_CHUNK(2); STAGE_B_CHUNK(3);
  STAGE_B_CHUNK(4); STAGE_B_CHUNK(5); STAGE_B_CHUNK(6); STAGE_B_CHUNK(7);
#undef STAGE_B_CHUNK

  // ---- Stage C: 8 rounds of wave-wide max extraction ----------------------
  int   idxs[KTOP];
  float vals[KTOP];

#pragma unroll
  for (int r = 0; r < KTOP; ++r) {
    float lmax = fmaxf(fmaxf(fmaxf(cmax[0], cmax[1]), fmaxf(cmax[2], cmax[3])),
                       fmaxf(fmaxf(cmax[4], cmax[5]), fmaxf(cmax[6], cmax[7])));
    float gm = lmax;
#pragma unroll
    for (int off = 16; off >= 1; off >>= 1) gm = fmaxf(gm, __shfl_xor(gm, off, 32));

    // smallest chunk in this lane whose max equals gm (chunks ascend in e).
    // Lanes with lmax < gm default to chunk 0, whose elements are all < gm,
    // so their scan below cannot match and me stays INT_MAX naturally.
    int kk = 0;
#pragma unroll
    for (int k = NCHUNK - 1; k >= 0; --k) kk = (cmax[k] == gm) ? k : kk;

    // rescan chunk kk (4 x b128): lowest matching global index, match count,
    // and the chunk max over strictly-smaller elements.
    int   me  = 0x7FFFFFFF;
    int   cnt = 0;
    float nms = NEG_INF;  // max over elements != gm
#pragma unroll
    for (int g = 0; g < 4; ++g) {
      const int    jj = 4 * kk + g;                  // group index (runtime)
      const float4 q  = l4[lane * NGRP + (jj ^ lane)];
#pragma unroll
      for (int t = 0; t < 4; ++t) {
        const float v  = (t == 0) ? q.x : (t == 1) ? q.y : (t == 2) ? q.z : q.w;
        const int   e  = 128 * jj + 4 * lane + t;    // global index in row
        const bool  eq = (v == gm);
        me  = min(me, eq ? e : 0x7FFFFFFF);
        cnt += eq ? 1 : 0;
        nms = fmaxf(nms, eq ? NEG_INF : v);
      }
    }
    // repaired chunk max after removing the first match: if a duplicate of gm
    // remains (cnt >= 2) the max is still gm (exact tie semantics).
    const float nm = (cnt >= 2) ? gm : nms;

    // lowest global index among all lanes holding gm -> exact top_k tie rule
    int mstar = me;
#pragma unroll
    for (int off = 16; off >= 1; off >>= 1) {
      const int o = __shfl_xor(mstar, off, 32);
      mstar = (o < mstar) ? o : mstar;
    }

    const bool owner = (me == mstar);  // unique: global indices are distinct
    if (owner) {
      // reconstruct swizzled LDS dword address of element mstar in this lane
      const int slot = lane * NGRP + ((mstar >> 7) ^ lane);
      lf[slot * 4 + (mstar & 3)] = NEG_INF;  // remove extracted element
    }
#pragma unroll
    for (int k = 0; k < NCHUNK; ++k)
      cmax[k] = (owner && (k == kk)) ? nm : cmax[k];

    idxs[r] = mstar;  // known wave-wide after the min reduction
    vals[r] = gm;
  }

  // ---- Stage D: sort the 8 (idx,val) pairs by index (Batcher OEMS-8) ------
#define CSWAP(a, b)                                                           \
  {                                                                           \
    const int   ia = idxs[a], ib = idxs[b];                                   \
    const float va = vals[a], vb = vals[b];                                   \
    const bool  sw = ia > ib;                                                 \
    idxs[a] = sw ? ib : ia;  idxs[b] = sw ? ia : ib;                          \
    vals[a] = sw ? vb : va;  vals[b] = sw ? va : vb;                          \
  }
  CSWAP(0, 1) CSWAP(2, 3) CSWAP(4, 5) CSWAP(6, 7)
  CSWAP(0, 2) CSWAP(1, 3) CSWAP(4, 6) CSWAP(5, 7)
  CSWAP(1, 2) CSWAP(5, 6)
  CSWAP(0, 4) CSWAP(1, 5) CSWAP(2, 6) CSWAP(3, 7)
  CSWAP(2, 4) CSWAP(3, 5)
  CSWAP(1, 2) CSWAP(3, 4) CSWAP(5, 6)
#undef CSWAP

  // ---- Stage E: lane 0 stores 8 floats as two b128 ------------------------
  if (lane == 0) {
    float4* op = (float4*)(out + (size_t)row * KTOP);
    op[0] = make_float4(vals[0], vals[1], vals[2], vals[3]);
    op[1] = make_float4(vals[4], vals[5], vals[6], vals[7]);
  }
}

extern "C" void kernel_launch(void* const* d_in, const int* in_sizes, int n_in,
                              void* d_out, int out_size, void* d_ws, size_t ws_size,
                              hipStream_t stream) {
  (void)n_in; (void)out_size; (void)d_ws; (void)ws_size;
  const float* x   = (const float*)d_in[0];
  float*       out = (float*)d_out;
  const int nrows = in_sizes[0] / LROW;  // 64*512 = 32768
  const int grid  = (nrows + WAVES - 1) / WAVES;
  kmax_pooling_kernel<<<grid, WAVES * 32, 0, stream>>>(x, out, nrows);
}